// HATM_28561532518900
// MI455X (gfx1250) — compile-verified
//
#include <hip/hip_runtime.h>
#include <stdint.h>

// ---------------------------------------------------------------------------
// Fused mask+transpose kernel for score(8,2048,4096) fp32 -> (student, teacher)
//   out[b][j][i] = masked[b][r][c],  j = c & 2047, i = 2r + (c>>11)
// One 512-thread workgroup owns 16 full rows staged in LDS via the CDNA5
// async global->LDS DMA; the hash-key radix-select runs while the DMA is in
// flight (ASYNCcnt waited only before the score-key select needs the data).
// ---------------------------------------------------------------------------

#define B_            8
#define R_            2048
#define C_            4096
#define ROWS_PER_WG   16
#define THREADS       512
#define K_FRONT       1228   // int(4096*0.30)
#define K_RAND        409    // int(4096*0.10)

// Padded LDS layout: addr(row, half, j) = row*SH_ROW + half*SH_HALF + j (floats)
#define SH_HALF       2052
#define SH_ROW        4104
#define SH_ROW_V4     1026   // SH_ROW/4 in 16B vectors

#define SMEM_WORDS    (ROWS_PER_WG * SH_ROW + ROWS_PER_WG * 256 + 4 * ROWS_PER_WG)
#define SMEM_BYTES    ((size_t)SMEM_WORDS * 4)

typedef float v4f __attribute__((ext_vector_type(4)));
// Exact pointee type expected by the async-LDS builtin (per hipcc diagnostic):
typedef int v4i __attribute__((vector_size(16)));
typedef __attribute__((address_space(1))) v4i gv4i_t;   // global 16B vector
typedef __attribute__((address_space(3))) v4i lv4i_t;   // LDS 16B vector

#if __has_builtin(__builtin_amdgcn_global_load_async_to_lds_b128)
#define HAS_ASYNC_LDS 1
#endif

__device__ __forceinline__ unsigned orderKey(float f) {
    unsigned u = __float_as_uint(f);
    return u ^ (unsigned)(((int)u >> 31) | 0x80000000);
}

// Deterministic per-(row,c) hash standing in for the threefry/argsort RNG stream.
__device__ __forceinline__ unsigned randKey(unsigned rowGlobal, unsigned c) {
    unsigned x = rowGlobal * 4096u + c;
    x ^= 0x9E3779B9u;
    x ^= x >> 16; x *= 0x7FEB352Du;
    x ^= x >> 15; x *= 0x846CA68Bu;
    x ^= x >> 16;
    return x;
}

// 4-pass 8-bit radix select: returns the key value at 0-indexed rank `want0`
// within the 4096 keys of local row w. mode 0: score keys from LDS.
// mode 1: hash keys (no LDS data dependency). One wave per row; hist/selP/selW
// regions are per-row, barriers keep the 16 waves phase-aligned. Bin scan is
// wave-parallel: 8 bins/lane + shfl_up exclusive scan, one lane resolves digit.
__device__ __forceinline__ unsigned radixSelect(
    int mode, int w, int lane, unsigned rowGlobal, unsigned want0,
    const float* vals, unsigned* hist, unsigned* selP, unsigned* selW) {
    if (lane == 0) { selP[w] = 0u; selW[w] = want0; }
    for (int pass = 0; pass < 4; ++pass) {
        const int shift = 24 - 8 * pass;
        for (int t = lane; t < 256; t += 32) hist[w * 256 + t] = 0u;
        __syncthreads();
        const unsigned prefix = selP[w];
#pragma unroll 4
        for (int c = lane; c < C_; c += 32) {
            unsigned key;
            if (mode == 0)
                key = orderKey(vals[w * SH_ROW + c + ((c >> 11) << 2)]);
            else
                key = randKey(rowGlobal, (unsigned)c);
            const bool match =
                (pass == 0) || ((key >> (shift + 8)) == (prefix >> (shift + 8)));
            if (match)
                atomicAdd(&hist[w * 256 + ((key >> shift) & 255u)], 1u);
        }
        __syncthreads();
        // Wave-parallel bin scan: lane owns bins [8*lane, 8*lane+8).
        unsigned binCnt[8];
        unsigned laneSum = 0u;
#pragma unroll
        for (int u = 0; u < 8; ++u) {
            binCnt[u] = hist[w * 256 + lane * 8 + u];
            laneSum += binCnt[u];
        }
        unsigned incl = laneSum;
#pragma unroll
        for (int d = 1; d < 32; d <<= 1) {
            unsigned t = __shfl_up(incl, d, 32);
            if (lane >= d) incl += t;
        }
        const unsigned excl = incl - laneSum;           // rank before my bins
        const unsigned want = selW[w];
        // Invariant: total matching > want, so exactly one lane holds the rank.
        if (excl <= want && want < excl + laneSum) {
            unsigned run = excl, digit = (unsigned)(lane * 8);
#pragma unroll
            for (int u = 0; u < 8; ++u) {
                if (run + binCnt[u] > want) { digit = (unsigned)(lane * 8 + u); break; }
                run += binCnt[u];
            }
            selP[w] = prefix | (digit << shift);
            selW[w] = want - run;
        }
        __syncthreads();
    }
    return selP[w];
}

__global__ void __launch_bounds__(THREADS, 1)
hatm_mask_transpose(const float* __restrict__ score,
                    float* __restrict__ outS,
                    float* __restrict__ outT) {
    extern __shared__ unsigned char smemRaw[];
    float*    vals    = (float*)smemRaw;                          // 16*4104 floats
    unsigned* hist    = (unsigned*)(vals + ROWS_PER_WG * SH_ROW); // 16*256
    unsigned* selP    = hist + ROWS_PER_WG * 256;                 // 16
    unsigned* selW    = selP + ROWS_PER_WG;                       // 16
    unsigned* thFront = selW + ROWS_PER_WG;                       // 16
    unsigned* thRand  = thFront + ROWS_PER_WG;                    // 16

    const int tid  = threadIdx.x;
    const int lane = tid & 31;
    const int w    = tid >> 5;                 // wave id == local row for selects
    const int b    = blockIdx.y;
    const int r0   = blockIdx.x * ROWS_PER_WG;

    const float* rowBase = score + ((size_t)b * R_ + (size_t)r0) * C_;

    // Address-space base pointers hoisted once (keeps the DMA loop tight).
    const gv4i_t* gBase = (const gv4i_t*)(uintptr_t)rowBase;
    lv4i_t*       lBase = (lv4i_t*)(uint32_t)(uintptr_t)vals;

    // ---- Phase A: launch async DMA of 16 rows (256 KiB) into LDS; no wait ----
    // 32 b128 DMAs per thread; unroll x8 keeps 8 issues statically in flight.
#pragma unroll 8
    for (int t = tid; t < ROWS_PER_WG * (C_ / 4); t += THREADS) {
        const int rloc = t >> 10;              // 1024 x 16B vectors per row
        const int k    = t & 1023;
        const int lvec = rloc * SH_ROW_V4 + k + (k >> 9);  // padded row/half layout
#if defined(HAS_ASYNC_LDS)
        __builtin_amdgcn_global_load_async_to_lds_b128(
            (gv4i_t*)(gBase + t), lBase + lvec, 0, 0);
#else
        v4f v = __builtin_nontemporal_load((const v4f*)(rowBase + (size_t)t * 4));
        *(v4f*)(vals + (size_t)lvec * 4) = v;
#endif
    }

    // ---- Phase B1 (overlapped with DMA): rank-409 select on hash keys --------
    const unsigned rowGlobal = (unsigned)(b * R_ + r0 + w);
    const unsigned tRand = radixSelect(1, w, lane, rowGlobal, K_RAND,
                                       vals, hist, selP, selW);
    if (lane == 0) thRand[w] = tRand;

    // ---- DMA completion: each wave drains its ASYNCcnt, then all rendezvous --
#if defined(HAS_ASYNC_LDS)
#if __has_builtin(__builtin_amdgcn_s_wait_asynccnt)
    __builtin_amdgcn_s_wait_asynccnt(0);
#else
    asm volatile("s_wait_asynccnt 0" ::: "memory");
#endif
#endif
    __syncthreads();

    // ---- Phase B2: rank-1228 select on order-preserving score keys -----------
    const unsigned tFront = radixSelect(0, w, lane, rowGlobal, K_FRONT,
                                        vals, hist, selP, selW);
    if (lane == 0) thFront[w] = tFront;
    __syncthreads();

    // ---- Phase C: masked, transposed, coalesced non-temporal stores ----------
    // thread -> 4 consecutive output i (2 rows x 2 halves) at one j; a wave
    // covers 32 consecutive i = four 128B contiguous segments per store.
    const int q     = tid & 7;        // i sub-block
    const int jslot = tid >> 3;       // 0..63
    const int iBase = 2 * r0 + 4 * q;
    float* outSb = outS + (size_t)b * R_ * C_;
    float* outTb = outT + (size_t)b * R_ * C_;

    for (int j = jslot; j < R_; j += 64) {
        float s4[4], t4[4];
#pragma unroll
        for (int m = 0; m < 4; ++m) {
            const int rloc = 2 * q + (m >> 1);
            const int half = m & 1;
            const int c    = j + (half << 11);
            const float v  = vals[rloc * SH_ROW + half * SH_HALF + j];
            const unsigned key = orderKey(v);
            const unsigned hk  = randKey((unsigned)(b * R_ + r0 + rloc), (unsigned)c);
            const bool front = key < thFront[rloc];
            const bool rnd   = hk  < thRand[rloc];
            s4[m] = (front || rnd) ? 0.0f : v;
            t4[m] = ((r0 + rloc) == 0) ? (front ? v : 0.0f) : v;  // teacher row-0 bug replicated
        }
        const size_t off = (size_t)j * C_ + (size_t)iBase;
        v4f sv = { s4[0], s4[1], s4[2], s4[3] };
        v4f tv = { t4[0], t4[1], t4[2], t4[3] };
        __builtin_nontemporal_store(sv, (v4f*)(outSb + off));
        __builtin_nontemporal_store(tv, (v4f*)(outTb + off));
    }
}

// ---------------------------------------------------------------------------

extern "C" void kernel_launch(void* const* d_in, const int* in_sizes, int n_in,
                              void* d_out, int out_size, void* d_ws, size_t ws_size,
                              hipStream_t stream) {
    (void)in_sizes; (void)n_in; (void)out_size; (void)d_ws; (void)ws_size;
    const float* score = (const float*)d_in[0];
    float* outS = (float*)d_out;
    float* outT = outS + (size_t)B_ * R_ * C_;

    // Dynamic LDS ~279 KB (CDNA5 allows up to 320 KB per workgroup).
    (void)hipFuncSetAttribute((const void*)hatm_mask_transpose,
                              hipFuncAttributeMaxDynamicSharedMemorySize,
                              (int)SMEM_BYTES);

    dim3 grid(R_ / ROWS_PER_WG, B_, 1);   // 128 x 8 = 1024 workgroups
    dim3 block(THREADS, 1, 1);
    hatm_mask_transpose<<<grid, block, SMEM_BYTES, stream>>>(score, outS, outT);
}